// SegmentGatingNetwork_70660801954255
// MI455X (gfx1250) — compile-verified
//
#include <hip/hip_runtime.h>
#include <math.h>

typedef float v2f __attribute__((ext_vector_type(2)));
typedef float v8f __attribute__((ext_vector_type(8)));

#define D_MODEL 1024
#define D_HID   256
#define N_EXP   64
#define TOK_PER_BLOCK 16

// D = A(16x4) * B(4x16) + C, fp32 throughout (exact-fidelity accumulation)
__device__ __forceinline__ v8f wmma_f32_4(v2f a, v2f b, v8f c) {
  return __builtin_amdgcn_wmma_f32_16x16x4_f32(false, a, false, b, (short)0, c,
                                               false, false);
}

__global__ __launch_bounds__(256) void moe_gate_fused(
    const float* __restrict__ x, const float* __restrict__ W1,
    const float* __restrict__ b1, const float* __restrict__ W2,
    const float* __restrict__ b2, float* __restrict__ gates,
    float* __restrict__ logits) {
  __shared__ __align__(16) float h_s[TOK_PER_BLOCK * D_HID];   // 16 KB
  __shared__ __align__(16) float lg_s[TOK_PER_BLOCK * N_EXP];  //  4 KB
  __shared__ float gval_s[TOK_PER_BLOCK][2];
  __shared__ int   gidx_s[TOK_PER_BLOCK][2];

  const int tid  = threadIdx.x;
  const int wave = tid >> 5;   // 8 waves (wave32)
  const int lane = tid & 31;
  const int ml   = lane & 15;  // M (A/C row) or N (B/C col) within a tile
  const int hi   = lane >> 4;  // K sub-block selector (+2)
  const int m0   = blockIdx.x * TOK_PER_BLOCK;

  // ---------------- GEMM1: h = tanh(x @ W1 + b1) -------------------------
  // Each wave computes h columns [c0, c0+32): two 16x16 accumulator tiles.
  const int c0 = wave * 32;
  const float* Ap = x + (size_t)(m0 + ml) * D_MODEL + hi * 2;
  v8f acc0 = {};
  v8f acc1 = {};
#pragma unroll 4
  for (int k = 0; k < D_MODEL; k += 4) {
    v2f a = *(const v2f*)(Ap + k);              // A[ml][kk], A[ml][kk+1]
    const int kk = k + hi * 2;
    const float* Bp = W1 + (size_t)kk * D_HID + c0 + ml;
    v2f bA, bB;
    bA.x = Bp[0];  bA.y = Bp[D_HID];            // W1[kk][c0+ml], W1[kk+1][c0+ml]
    bB.x = Bp[16]; bB.y = Bp[D_HID + 16];
    acc0 = wmma_f32_4(a, bA, acc0);
    acc1 = wmma_f32_4(a, bB, acc1);
  }
  {
    const float bias0 = b1[c0 + ml];
    const float bias1 = b1[c0 + 16 + ml];
#pragma unroll
    for (int r = 0; r < 8; ++r) {
      const int M = r + hi * 8;                 // C/D row layout
      h_s[M * D_HID + c0 + ml]      = tanhf(acc0[r] + bias0);
      h_s[M * D_HID + c0 + 16 + ml] = tanhf(acc1[r] + bias1);
    }
  }
  __syncthreads();

  // ---------------- GEMM2: logits = h @ W2 + b2 --------------------------
  if (wave < 4) {                               // whole-wave uniform branch
    const int e0 = wave * 16;
    v8f acc = {};
#pragma unroll 4
    for (int k = 0; k < D_HID; k += 4) {
      const int kk = k + hi * 2;
      v2f a = *(const v2f*)&h_s[ml * D_HID + kk];
      v2f bf;
      bf.x = W2[kk * N_EXP + e0 + ml];
      bf.y = W2[(kk + 1) * N_EXP + e0 + ml];
      acc = wmma_f32_4(a, bf, acc);
    }
    const float bias = b2[e0 + ml];
#pragma unroll
    for (int r = 0; r < 8; ++r) {
      const int M = r + hi * 8;
      const float v = acc[r] + bias;
      logits[(size_t)(m0 + M) * N_EXP + e0 + ml] = v;
      lg_s[M * N_EXP + e0 + ml] = v;
    }
  }
  __syncthreads();

  // ---------------- top-2 + softmax over the 2 selected logits -----------
  if (tid < TOK_PER_BLOCK) {
    float m1 = -INFINITY, m2 = -INFINITY;
    int i1 = 0, i2 = 0;
    for (int e = 0; e < N_EXP; ++e) {
      const float v = lg_s[tid * N_EXP + e];
      if (v > m1)      { m2 = m1; i2 = i1; m1 = v; i1 = e; }
      else if (v > m2) { m2 = v; i2 = e; }
    }
    const float ex  = __expf(m2 - m1);          // stable: m1 >= m2
    const float inv = 1.0f / (1.0f + ex);
    gval_s[tid][0] = inv;
    gval_s[tid][1] = ex * inv;
    gidx_s[tid][0] = i1;
    gidx_s[tid][1] = i2;
  }
  __syncthreads();

  // ---------------- dense scatter: 4 gates/thread, coalesced float4 ------
  {
    const int t  = tid >> 4;                    // token within block
    const int eb = (tid & 15) * 4;              // first expert of this quad
    const int i1 = gidx_s[t][0], i2 = gidx_s[t][1];
    const float g1 = gval_s[t][0], g2 = gval_s[t][1];
    float o[4];
#pragma unroll
    for (int j = 0; j < 4; ++j) {
      const int e = eb + j;
      o[j] = (e == i1) ? g1 : ((e == i2) ? g2 : 0.0f);
    }
    float4 ov = make_float4(o[0], o[1], o[2], o[3]);
    *(float4*)&gates[(size_t)(m0 + t) * N_EXP + eb] = ov;
  }
}

extern "C" void kernel_launch(void* const* d_in, const int* in_sizes, int n_in,
                              void* d_out, int out_size, void* d_ws,
                              size_t ws_size, hipStream_t stream) {
  const float* x  = (const float*)d_in[0];
  const float* W1 = (const float*)d_in[1];
  const float* b1 = (const float*)d_in[2];
  const float* W2 = (const float*)d_in[3];
  const float* b2 = (const float*)d_in[4];

  const int n_tok = in_sizes[0] / D_MODEL;      // 4*4096 = 16384 tokens
  float* gates  = (float*)d_out;                              // [n_tok, 64]
  float* logits = (float*)d_out + (size_t)n_tok * N_EXP;      // [n_tok, 64]

  const int blocks = n_tok / TOK_PER_BLOCK;     // 1024
  moe_gate_fused<<<blocks, 256, 0, stream>>>(x, W1, b1, W2, b2, gates, logits);
}